// TorchPairDistances_72378788872234
// MI455X (gfx1250) — compile-verified
//
#include <hip/hip_runtime.h>
#include <math.h>

#define N_ELK   4
#define BATCH   128
#define NATOM   128
#define NCONN   256
#define F_ATOM  176
#define F_PAIR  32
#define FIN1    384
#define FOUT1   128
#define FOUT2   96
#define FOUT3   2
#define NPAIR   (BATCH * NCONN)   /* 32768 */
#define NEXP    16

#define STR1 388   /* LDS row stride (floats) for x (384) : 4r mod 64 distinct banks */
#define STR2 132   /* for h1 (128) */
#define STR3 100   /* for h2 (96) */

typedef __attribute__((ext_vector_type(2))) float v2f;
typedef __attribute__((ext_vector_type(8))) float v8f;

__device__ __forceinline__ float celu1(float x) {
    return x > 0.0f ? x : expm1f(x);
}

// ---------------- bucketing kernels ----------------

__global__ void k_init(int* counts) {
    if (threadIdx.x < NEXP) counts[threadIdx.x] = 0;
}

__global__ void k_count(const int* __restrict__ conn, const int* __restrict__ el,
                        int* __restrict__ counts, float* __restrict__ out) {
    int p = blockIdx.x * blockDim.x + threadIdx.x;
    if (p >= NPAIR) return;
    int a0 = conn[2 * p];
    if (a0 < 0) {                       // padded pair -> zero output, no expert
        out[2 * p] = 0.0f;
        out[2 * p + 1] = 0.0f;
        return;
    }
    int a1 = conn[2 * p + 1];
    int b  = p >> 8;                    // p / NCONN
    int e0 = el[b * NATOM + a0];
    int e1 = el[b * NATOM + a1];
    atomicAdd(&counts[e0 * N_ELK + e1], 1);
}

__global__ void k_prefix(const int* __restrict__ counts,
                         int* __restrict__ offsets, int* __restrict__ cursors) {
    if (threadIdx.x == 0) {
        int run = 0;
        for (int i = 0; i < NEXP; ++i) {
            offsets[i] = run;
            cursors[i] = run;
            run += counts[i];
        }
    }
}

__global__ void k_scatter(const int* __restrict__ conn, const int* __restrict__ el,
                          int* __restrict__ cursors, int* __restrict__ bucket) {
    int p = blockIdx.x * blockDim.x + threadIdx.x;
    if (p >= NPAIR) return;
    int a0 = conn[2 * p];
    if (a0 < 0) return;
    int a1 = conn[2 * p + 1];
    int b  = p >> 8;
    int e0 = el[b * NATOM + a0];
    int e1 = el[b * NATOM + a1];
    int pos = atomicAdd(&cursors[e0 * N_ELK + e1], 1);
    bucket[pos] = p;
}

// ---------------- MoE GEMM kernel: one wave32 per (expert, 16-row tile) -------------

__global__ __launch_bounds__(32, 8)
void k_moe(const int* __restrict__ conn,
           const float* __restrict__ sym, const float* __restrict__ pairf,
           const float* __restrict__ W1, const float* __restrict__ B1,
           const float* __restrict__ W2, const float* __restrict__ B2,
           const float* __restrict__ W3, const float* __restrict__ B3,
           const int* __restrict__ offsets, const int* __restrict__ counts,
           const int* __restrict__ bucket, float* __restrict__ out) {
    __shared__ float xs[16 * STR1];   // 24832 B
    __shared__ float h1[16 * STR2];   //  8448 B
    __shared__ float h2[16 * STR3];   //  6400 B
    __shared__ int s_p[16];
    __shared__ int s_b0[16];
    __shared__ int s_b1[16];

    const int e    = blockIdx.y;
    const int row0 = blockIdx.x * 16;
    const int cnt  = counts[e];
    if (row0 >= cnt) return;                 // wave-uniform exit
    const int start = offsets[e];
    const int nval  = min(16, cnt - row0);

    const int lane = threadIdx.x;
    const int rlo  = lane & 15;
    const int koff = (lane >> 4) << 1;       // 0 or 2: K sub-offset per A/B layout
    const int mbase = (lane >> 4) * 8;       // C/D row base per lane half

    // ---- per-row metadata (lanes 0..15) ----
    if (lane < 16) {
        int p = -1, b0 = 0, b1 = 0;
        if (lane < nval) {
            p = bucket[start + row0 + lane];
            int b  = p >> 8;
            int a0 = conn[2 * p];
            int a1 = conn[2 * p + 1];
            b0 = (b * NATOM + a0) * F_ATOM;
            b1 = (b * NATOM + a1) * F_ATOM;
        }
        s_p[lane] = p; s_b0[lane] = b0; s_b1[lane] = b1;
    }
    __syncthreads();

    // ---- gather x = [fa(176) | fb(176) | pf(32)] into LDS as float4 ----
    for (int r = 0; r < 16; ++r) {
        const int p = s_p[r];
        const float4* f0 = (const float4*)(sym + s_b0[r]);
        const float4* f1 = (const float4*)(sym + s_b1[r]);
        const float4* fp = (const float4*)(pairf + (p < 0 ? 0 : p) * F_PAIR);
        for (int q = lane; q < 96; q += 32) {   // 96 float4 = 384 floats
            float4 v;
            if (p < 0)        v = make_float4(0.f, 0.f, 0.f, 0.f);
            else if (q < 44)  v = f0[q];
            else if (q < 88)  v = f1[q - 44];
            else              v = fp[q - 88];
            *(float4*)&xs[r * STR1 + q * 4] = v;
        }
    }
    __syncthreads();

    // ---- layer 1: [16,384] @ [384,128] + b1, celu.  FP32 WMMA 16x16x4. ----
    {
        v8f acc[8];
#pragma unroll
        for (int nt = 0; nt < 8; ++nt) {
            float bv = B1[e * FOUT1 + nt * 16 + rlo];
            acc[nt] = (v8f){bv, bv, bv, bv, bv, bv, bv, bv};
        }
        const float* xrow = &xs[rlo * STR1];
        for (int kb = 0; kb < FIN1; kb += 4) {
            v2f a = *(const v2f*)&xrow[kb + koff];
            const float* wp = W1 + ((size_t)(e * FIN1 + kb + koff)) * FOUT1 + rlo;
#pragma unroll
            for (int nt = 0; nt < 8; ++nt) {
                v2f bf;
                bf.x = wp[nt * 16];
                bf.y = wp[nt * 16 + FOUT1];
                acc[nt] = __builtin_amdgcn_wmma_f32_16x16x4_f32(
                    false, a, false, bf, (short)0, acc[nt], false, false);
            }
        }
#pragma unroll
        for (int nt = 0; nt < 8; ++nt)
#pragma unroll
            for (int v = 0; v < 8; ++v)
                h1[(mbase + v) * STR2 + nt * 16 + rlo] = celu1(acc[nt][v]);
    }
    __syncthreads();

    // ---- layer 2: [16,128] @ [128,96] + b2, celu ----
    {
        v8f acc[6];
#pragma unroll
        for (int nt = 0; nt < 6; ++nt) {
            float bv = B2[e * FOUT2 + nt * 16 + rlo];
            acc[nt] = (v8f){bv, bv, bv, bv, bv, bv, bv, bv};
        }
        const float* xrow = &h1[rlo * STR2];
        for (int kb = 0; kb < FOUT1; kb += 4) {
            v2f a = *(const v2f*)&xrow[kb + koff];
            const float* wp = W2 + ((size_t)(e * FOUT1 + kb + koff)) * FOUT2 + rlo;
#pragma unroll
            for (int nt = 0; nt < 6; ++nt) {
                v2f bf;
                bf.x = wp[nt * 16];
                bf.y = wp[nt * 16 + FOUT2];
                acc[nt] = __builtin_amdgcn_wmma_f32_16x16x4_f32(
                    false, a, false, bf, (short)0, acc[nt], false, false);
            }
        }
#pragma unroll
        for (int nt = 0; nt < 6; ++nt)
#pragma unroll
            for (int v = 0; v < 8; ++v)
                h2[(mbase + v) * STR3 + nt * 16 + rlo] = celu1(acc[nt][v]);
    }
    __syncthreads();

    // ---- layer 3: [16,96] @ [96,2] + b3 (VALU; 16 rows x 2 cols = 32 lanes) ----
    {
        const int row = lane >> 1;
        const int oc  = lane & 1;
        float sum = B3[e * FOUT3 + oc];
        const float* xr = &h2[row * STR3];
        const float* w3 = W3 + (size_t)e * FOUT2 * FOUT3 + oc;
        for (int k = 0; k < FOUT2; ++k)
            sum += xr[k] * w3[k * FOUT3];
        if (row < nval) {
            int p = s_p[row];
            out[p * 2 + oc] = sum;
        }
    }
}

// ---------------- launcher ----------------

extern "C" void kernel_launch(void* const* d_in, const int* in_sizes, int n_in,
                              void* d_out, int out_size, void* d_ws, size_t ws_size,
                              hipStream_t stream) {
    const int*   elements = (const int*)d_in[0];
    const int*   conn     = (const int*)d_in[1];
    const float* sym      = (const float*)d_in[2];
    const float* pairf    = (const float*)d_in[3];
    const float* W1       = (const float*)d_in[4];
    const float* B1       = (const float*)d_in[5];
    const float* W2       = (const float*)d_in[6];
    const float* B2       = (const float*)d_in[7];
    const float* W3       = (const float*)d_in[8];
    const float* B3       = (const float*)d_in[9];
    float* out = (float*)d_out;

    int* ws      = (int*)d_ws;
    int* counts  = ws;        // 16
    int* offsets = ws + 16;   // 16
    int* cursors = ws + 32;   // 16
    int* bucket  = ws + 64;   // NPAIR

    k_init<<<1, 32, 0, stream>>>(counts);
    k_count<<<NPAIR / 256, 256, 0, stream>>>(conn, elements, counts, out);
    k_prefix<<<1, 32, 0, stream>>>(counts, offsets, cursors);
    k_scatter<<<NPAIR / 256, 256, 0, stream>>>(conn, elements, cursors, bucket);

    dim3 grid(NPAIR / 16, NEXP, 1);   // worst case: all pairs on one expert
    k_moe<<<grid, 32, 0, stream>>>(conn, sym, pairf, W1, B1, W2, B2, W3, B3,
                                   offsets, counts, bucket, out);
}